// BaseModel_77429670412293
// MI455X (gfx1250) — compile-verified
//
#include <hip/hip_runtime.h>
#include <cstdint>

// Problem constants (match the reference)
#define N_USERS 200000
#define N_ITEMS 100000
#define EMB 64
#define N_NODES (N_USERS + N_ITEMS)          // 300,000
#define NN (N_NODES * EMB)                   // 19,200,000 floats per matrix
#define NV4 (NN / 4)                         // 4,800,000 float4
#define USER_V4 (N_USERS * EMB / 4)          // 3,200,000 float4

// ---------------------------------------------------------------------------
// k_init: x_a = concat(user, item); out = x_a (layer-0 term); x_b = 0
// ---------------------------------------------------------------------------
__global__ void __launch_bounds__(256) k_init(const float4* __restrict__ eu,
                                              const float4* __restrict__ ei,
                                              float4* __restrict__ xa,
                                              float4* __restrict__ out4,
                                              float4* __restrict__ xb) {
    int i = blockIdx.x * blockDim.x + threadIdx.x;
    if (i >= NV4) return;
    float4 v = (i < USER_V4) ? eu[i] : ei[i - USER_V4];
    xa[i]   = v;
    out4[i] = v;
    xb[i]   = make_float4(0.f, 0.f, 0.f, 0.f);
}

// ---------------------------------------------------------------------------
// k_acc: out += xn (add this layer's embedding); zero zbuf (next SpMM dest)
// ---------------------------------------------------------------------------
__global__ void __launch_bounds__(256) k_acc(float4* __restrict__ out4,
                                             const float4* __restrict__ xn,
                                             float4* __restrict__ zbuf) {
    int i = blockIdx.x * blockDim.x + threadIdx.x;
    if (i >= NV4) return;
    float4 o = out4[i];
    float4 x = xn[i];
    o.x += x.x; o.y += x.y; o.z += x.z; o.w += x.w;
    out4[i] = o;
    zbuf[i] = make_float4(0.f, 0.f, 0.f, 0.f);
}

// ---------------------------------------------------------------------------
// k_final: out = (out + xn) * 0.25  (mean over the 4 layer matrices)
// ---------------------------------------------------------------------------
__global__ void __launch_bounds__(256) k_final(float4* __restrict__ out4,
                                               const float4* __restrict__ xn) {
    int i = blockIdx.x * blockDim.x + threadIdx.x;
    if (i >= NV4) return;
    float4 o = out4[i];
    float4 x = xn[i];
    o.x = (o.x + x.x) * 0.25f;
    o.y = (o.y + x.y) * 0.25f;
    o.z = (o.z + x.z) * 0.25f;
    o.w = (o.w + x.w) * 0.25f;
    out4[i] = o;
}

// ---------------------------------------------------------------------------
// k_spmm: y[r] += val * x[c] for each edge (r, c).
//
// One wave32 processes 2 edges per step (16 lanes x float4 = 256B per row).
// Two-stage software pipeline, branch-free in steady state:
//   - 256B row gathers run on the CDNA5 async copy engine
//     (global_load_async_to_lds_b128, ASYNCcnt) into a per-wave LDS double
//     buffer: the gather for pair p+stride overlaps the scale+scatter of p.
//   - Edge scalars (col/row/val) are prefetched one full iteration ahead and
//     carried in registers, so the gather address never stalls on ecol[].
//   - Tail handling: edge indices clamped (always-valid addresses), val
//     masked to 0; the final extra gather is drained by s_endpgm wait-idle.
// Scatter uses no-return global_atomic_add_f32; both src and dst matrices
// (76.8MB each) are L2-resident (192MB L2), so the random traffic stays on-die.
// ---------------------------------------------------------------------------
__global__ void __launch_bounds__(256) k_spmm(const float* __restrict__ x,
                                              float* __restrict__ y,
                                              const float* __restrict__ eval_,
                                              const int* __restrict__ erow,
                                              const int* __restrict__ ecol,
                                              int nEdges) {
    __shared__ float4 stage[2][256];   // 8 waves * 32 lanes, double buffered

    const int lane = threadIdx.x & 31;
    const int wib  = threadIdx.x >> 5;          // wave index in block
    const int sub  = lane & 15;                 // float4 slot within edge row
    const int half = lane >> 4;                 // which edge of the pair
    const int nPairs = (nEdges + 1) >> 1;
    const int nWaves = gridDim.x * (blockDim.x >> 5);
    int p = blockIdx.x * (blockDim.x >> 5) + wib;
    if (p >= nPairs) return;

    float4* slot0 = &stage[0][wib * 32 + lane];
    float4* slot1 = &stage[1][wib * 32 + lane];
    const uint32_t ldsA0 = (uint32_t)(uintptr_t)slot0;  // LDS byte address
    const uint32_t ldsA1 = (uint32_t)(uintptr_t)slot1;
    const int lastE = nEdges - 1;

    // Prefetch edge scalars for a pair (clamped-safe; val masked on tail).
    auto fetchScalars = [&](int q, int& c, int& r, float& v) {
        int eRaw = q * 2 + half;
        int e = (eRaw > lastE) ? lastE : eRaw;
        c = ecol[e];
        r = erow[e];
        float t = eval_[e];
        v = (eRaw > lastE) ? 0.0f : t;
    };

    // Async-gather one edge pair's rows (2 x 256B) into an LDS buffer.
    auto issue = [&](int c, uint32_t ldsAddr) {
        const float* g = x + (size_t)c * EMB + sub * 4;
        // The recycled buffer's previous ds_load must retire before the async
        // engine may overwrite it (DScnt and ASYNCcnt are independent).
        asm volatile("s_wait_dscnt 0" ::: "memory");
        asm volatile("global_load_async_to_lds_b128 %0, %1, off"
                     :: "v"(ldsAddr), "v"((uint64_t)(uintptr_t)g)
                     : "memory");
    };

    // ---- pipeline prologue -------------------------------------------------
    int   c0, r, rn;
    float v, vn;
    fetchScalars(p, c0, r, v);              // scalars for pair p
    issue(c0, ldsA0);                       // gather(p) -> buf0
    int cn;
    fetchScalars(p + nWaves, cn, rn, vn);   // scalars for pair p+stride

    // ---- steady-state loop -------------------------------------------------
    int parity = 0;
    while (true) {
        // Gather for pair p+stride into the other buffer (clamped-safe even
        // past the end; at most one wasted 512B gather per wave).
        issue(cn, parity ? ldsA0 : ldsA1);

        // Prefetch scalars two pairs ahead (one-iteration slack).
        int c2, r2; float v2;
        fetchScalars(p + 2 * nWaves, c2, r2, v2);

        // Async ops complete in order: <=1 outstanding => gather(p) landed.
        asm volatile("s_wait_asynccnt 0x1" ::: "memory");
        float4 d = parity ? *slot1 : *slot0;          // ds_load_b128

        float* o = y + (size_t)r * EMB + sub * 4;
        unsafeAtomicAdd(o + 0, d.x * v);              // global_atomic_add_f32
        unsafeAtomicAdd(o + 1, d.y * v);
        unsafeAtomicAdd(o + 2, d.z * v);
        unsafeAtomicAdd(o + 3, d.w * v);

        p += nWaves;
        if (p >= nPairs) break;                        // uniform per wave
        r = rn; v = vn; cn = c2; rn = r2; vn = v2;
        parity ^= 1;
    }
    // s_endpgm performs an implicit wait-idle, draining the last async gather.
}

// ---------------------------------------------------------------------------
// Host launcher. Inputs (reference dict order):
//   d_in[0]=emb_user f32[200000*64], d_in[1]=emb_item f32[100000*64],
//   d_in[2]=edge_val f32[5M], d_in[3]=edge_row i32[5M], d_in[4]=edge_col i32[5M]
// d_out = agg f32[300000*64] (users then items, matching concat order).
// d_ws: two 76.8MB ping-pong matrices.
// ---------------------------------------------------------------------------
extern "C" void kernel_launch(void* const* d_in, const int* in_sizes, int n_in,
                              void* d_out, int out_size, void* d_ws, size_t ws_size,
                              hipStream_t stream) {
    const float* eu = (const float*)d_in[0];
    const float* ei = (const float*)d_in[1];
    const float* ev = (const float*)d_in[2];
    const int*   er = (const int*)d_in[3];
    const int*   ec = (const int*)d_in[4];
    const int nEdges = in_sizes[2];

    float* out = (float*)d_out;
    float* xa  = (float*)d_ws;          // 19.2M floats
    float* xb  = xa + NN;               // 19.2M floats

    const dim3 eb(256), eg((NV4 + 255) / 256);
    const dim3 sb(256), sg(4096);       // 32768 waves, ~76 edge pairs each

    // out = x0; xa = x0; xb = 0
    k_init<<<eg, eb, 0, stream>>>((const float4*)eu, (const float4*)ei,
                                  (float4*)xa, (float4*)out, (float4*)xb);
    // layer 1: xb = A*xa;  out += xb; xa = 0
    k_spmm<<<sg, sb, 0, stream>>>(xa, xb, ev, er, ec, nEdges);
    k_acc <<<eg, eb, 0, stream>>>((float4*)out, (const float4*)xb, (float4*)xa);
    // layer 2: xa = A*xb;  out += xa; xb = 0
    k_spmm<<<sg, sb, 0, stream>>>(xb, xa, ev, er, ec, nEdges);
    k_acc <<<eg, eb, 0, stream>>>((float4*)out, (const float4*)xa, (float4*)xb);
    // layer 3: xb = A*xa;  out = (out + xb) / 4
    k_spmm<<<sg, sb, 0, stream>>>(xa, xb, ev, er, ec, nEdges);
    k_final<<<eg, eb, 0, stream>>>((float4*)out, (const float4*)xb);
}